// PoseLoss_90555090468868
// MI455X (gfx1250) — compile-verified
//
#include <hip/hip_runtime.h>
#include <hip/hip_bf16.h>
#include <math.h>

// -------- CDNA5 WMMA types (wave32): 16x16x4 f32 => A/B = 2 VGPRs, C/D = 8 VGPRs
typedef __attribute__((ext_vector_type(2))) float v2f;
typedef __attribute__((ext_vector_type(8))) float v8f;

// Wave-level (32-lane) sum using V_WMMA_F32_16X16X4_F32.
// A = ones(16x4). B: VGPR0 = per-lane partial, VGPR1 = 0, so
//   B[0,n] = p[n] (lanes 0-15), B[2,n] = p[n+16] (lanes 16-31), rows 1,3 = 0.
// D[m,n] = sum_k A[m,k]*B[k,n] = p[n] + p[n+16]  -> every lane's c[0] holds
// p[lane%16] + p[lane%16 + 16]. Four xor-shuffles finish the 16-value sum.
// Requires EXEC == all ones at the call site (ISA 7.12 restriction).
__device__ __forceinline__ float wave_sum_wmma(float v) {
  v2f a; a[0] = 1.0f; a[1] = 1.0f;
  v2f b; b[0] = v;    b[1] = 0.0f;
  v8f c = {};
  c = __builtin_amdgcn_wmma_f32_16x16x4_f32(
      /*neg_a=*/false, a, /*neg_b=*/false, b,
      /*c_mod=*/(short)0, c, /*reuse_a=*/false, /*reuse_b=*/false);
  float r = c[0];
  r += __shfl_xor(r, 1);
  r += __shfl_xor(r, 2);
  r += __shfl_xor(r, 4);
  r += __shfl_xor(r, 8);
  return r;  // full 32-lane sum, replicated to all lanes
}

__device__ __forceinline__ float branch_loss(float x0, float x1, float x2,
                                             float4 q,
                                             float gx, float gy, float gz,
                                             float nw, float nx, float ny, float nz,
                                             float w_q) {
  float t0 = x0 - gx, t1 = x1 - gy, t2 = x2 - gz;
  float dp = sqrtf(t0 * t0 + t1 * t1 + t2 * t2);
  float u0 = q.x - nw, u1 = q.y - nx, u2 = q.z - ny, u3 = q.w - nz;
  float dq = sqrtf(u0 * u0 + u1 * u1 + u2 * u2 + u3 * u3);
  return dp + w_q * dq;
}

__global__ void __launch_bounds__(256)
pose_loss_partial(const float* __restrict__ p1_xyz,
                  const float* __restrict__ p1_wpqr,
                  const float* __restrict__ p2_xyz,
                  const float* __restrict__ p2_wpqr,
                  const float* __restrict__ p3_xyz,
                  const float* __restrict__ p3_wpqr,
                  const float* __restrict__ poseGT,
                  float* __restrict__ partials,
                  int n) {
  const int tid = blockIdx.x * blockDim.x + threadIdx.x;
  const int stride = gridDim.x * blockDim.x;

  float acc = 0.0f;
  for (int i = tid; i < n; i += stride) {
    // ---- ground truth pose (7 floats, rows only dword-aligned) ----
    const float* g = poseGT + 7 * (size_t)i;
    const float gx = g[0], gy = g[1], gz = g[2];
    const float qw = g[3], qx = g[4], qy = g[5], qz = g[6];
    const float qs  = qw * qw + qx * qx + qy * qy + qz * qz;
    const float inv = 1.0f / fmaxf(sqrtf(qs), 1e-12f);
    const float nw = qw * inv, nx = qx * inv, ny = qy * inv, nz = qz * inv;

    // ---- predictions: wpqr rows are 16B-aligned -> b128 loads ----
    const float4 a1 = ((const float4*)p1_wpqr)[i];
    const float4 a2 = ((const float4*)p2_wpqr)[i];
    const float4 a3 = ((const float4*)p3_wpqr)[i];
    const float* x1 = p1_xyz + 3 * (size_t)i;
    const float* x2 = p2_xyz + 3 * (size_t)i;
    const float* x3 = p3_xyz + 3 * (size_t)i;

    const float loss1 = branch_loss(x1[0], x1[1], x1[2], a1, gx, gy, gz, nw, nx, ny, nz, 150.0f);
    const float loss2 = branch_loss(x2[0], x2[1], x2[2], a2, gx, gy, gz, nw, nx, ny, nz, 150.0f);
    const float loss3 = branch_loss(x3[0], x3[1], x3[2], a3, gx, gy, gz, nw, nx, ny, nz, 500.0f);

    acc += 0.3f * loss1 + 0.3f * loss2 + 1.0f * loss3;
  }

  // ---- deterministic block reduction: wave (WMMA) -> LDS -> lane 0 ----
  __shared__ float swave[8];  // 256 threads = 8 wave32
  const float wsum = wave_sum_wmma(acc);  // all threads reconverged here
  const int lane = threadIdx.x & 31;
  const int wid  = threadIdx.x >> 5;
  if (lane == 0) swave[wid] = wsum;
  __syncthreads();
  if (threadIdx.x == 0) {
    float s = 0.0f;
#pragma unroll
    for (int j = 0; j < 8; ++j) s += swave[j];
    partials[blockIdx.x] = s;
  }
}

__global__ void __launch_bounds__(256)
pose_loss_final(const float* __restrict__ partials, int nparts,
                float* __restrict__ out, float inv_n) {
  float acc = 0.0f;
  for (int i = threadIdx.x; i < nparts; i += blockDim.x) acc += partials[i];

  __shared__ float swave[8];
  const float wsum = wave_sum_wmma(acc);
  const int lane = threadIdx.x & 31;
  const int wid  = threadIdx.x >> 5;
  if (lane == 0) swave[wid] = wsum;
  __syncthreads();
  if (threadIdx.x == 0) {
    float s = 0.0f;
#pragma unroll
    for (int j = 0; j < 8; ++j) s += swave[j];
    out[0] = s * inv_n;
  }
}

extern "C" void kernel_launch(void* const* d_in, const int* in_sizes, int n_in,
                              void* d_out, int out_size, void* d_ws, size_t ws_size,
                              hipStream_t stream) {
  const float* p1_xyz  = (const float*)d_in[0];
  const float* p1_wpqr = (const float*)d_in[1];
  const float* p2_xyz  = (const float*)d_in[2];
  const float* p2_wpqr = (const float*)d_in[3];
  const float* p3_xyz  = (const float*)d_in[4];
  const float* p3_wpqr = (const float*)d_in[5];
  const float* poseGT  = (const float*)d_in[6];

  const int n = in_sizes[6] / 7;  // B

  // Block partials live in workspace (blocks * 4 bytes).
  float* partials = (float*)d_ws;
  int blocks = 1024;  // 1024 blocks x 256 thr = 2^18 lanes -> 4 samples/lane at B=2^20
  const int max_blocks = (int)(ws_size / sizeof(float));
  if (blocks > max_blocks) blocks = max_blocks > 0 ? max_blocks : 1;

  pose_loss_partial<<<blocks, 256, 0, stream>>>(p1_xyz, p1_wpqr, p2_xyz, p2_wpqr,
                                                p3_xyz, p3_wpqr, poseGT, partials, n);
  pose_loss_final<<<1, 256, 0, stream>>>(partials, blocks, (float*)d_out,
                                         1.0f / (float)n);
}